// Net_60739427500576
// MI455X (gfx1250) — compile-verified
//
#include <hip/hip_runtime.h>
#include <hip/hip_bf16.h>

// ---------------------------------------------------------------------------
// 2-layer GCN (1 -> 8 -> 1 channels), B=8 batches, algebraically collapsed:
//   agg[b,n] = sum_{e: dst=n} dis[s]*dis[n]*x[b,s]  + dis[n]^2 * x[b,n]
//   y[b,n]   = sum_c relu(agg[b,n]*W1[c]+b1[c]) * W2[c]     (WMMA f32 16x16x4)
//   out[b,n] = sigmoid( sum_{e: dst=n} dis[s]*dis[n]*y[b,s] + dis[n]^2*y[b,n] + b2 )
// Atomic-f32 bound (~26M global_atomic_add_f32); [N][8] layout makes each
// edge's 8 atomics land in one 32B/128B region and keeps gathers L2-local.
// ---------------------------------------------------------------------------

typedef float v2f_t __attribute__((ext_vector_type(2)));
typedef float v8f_t __attribute__((ext_vector_type(8)));

#define BATCH 8

__device__ __forceinline__ void atomic_add_f32(float* p, float v) {
  // native global_atomic_add_f32 (no CAS loop)
  unsafeAtomicAdd(p, v);
}

// --- zero a float region (grid-stride) -------------------------------------
__global__ void k_zero(float* p, long long n) {
  long long i = (long long)blockIdx.x * blockDim.x + threadIdx.x;
  long long stride = (long long)gridDim.x * blockDim.x;
  for (; i < n; i += stride) p[i] = 0.0f;
}

// --- transpose x[b,n] -> xt[n*8+b] -----------------------------------------
__global__ void k_transpose(const float* __restrict__ x, float* __restrict__ xt, int N) {
  int i = blockIdx.x * blockDim.x + threadIdx.x;
  if (i >= N * BATCH) return;
  int n = i >> 3, b = i & 7;
  xt[i] = x[(long long)b * N + n];
}

// --- degree: deg[dst] += 1 --------------------------------------------------
__global__ void k_deg(const long long* __restrict__ ei, float* __restrict__ deg, int E) {
  int e = blockIdx.x * blockDim.x + threadIdx.x;
  if (e >= E) return;
  int d = (int)ei[E + e];
  atomic_add_f32(&deg[d], 1.0f);
}

// --- dis[n] = rsqrt(deg[n] + 1)  (self-loop) -------------------------------
__global__ void k_rsqrt(const float* __restrict__ deg, float* __restrict__ dis, int N) {
  int n = blockIdx.x * blockDim.x + threadIdx.x;
  if (n >= N) return;
  dis[n] = rsqrtf(deg[n] + 1.0f);
}

// --- scatter pass: acc[dst*8+b] += dis[s]*dis[d]*feat[src*8+b] -------------
__global__ void k_edge_scatter(const long long* __restrict__ ei,
                               const float* __restrict__ dis,
                               const float* __restrict__ feat,
                               float* __restrict__ acc, int E) {
  int e = blockIdx.x * blockDim.x + threadIdx.x;
  if (e >= E) return;
  int s = (int)ei[e];
  int d = (int)ei[E + e];
  float w = dis[s] * dis[d];
  const float4* sf = (const float4*)(feat + (long long)s * BATCH);
  float4 f0 = sf[0], f1 = sf[1];
  float* dp = acc + (long long)d * BATCH;
  atomic_add_f32(dp + 0, w * f0.x);
  atomic_add_f32(dp + 1, w * f0.y);
  atomic_add_f32(dp + 2, w * f0.z);
  atomic_add_f32(dp + 3, w * f0.w);
  atomic_add_f32(dp + 4, w * f1.x);
  atomic_add_f32(dp + 5, w * f1.y);
  atomic_add_f32(dp + 6, w * f1.z);
  atomic_add_f32(dp + 7, w * f1.w);
}

// --- per-node MLP via V_WMMA_F32_16X16X4_F32 -------------------------------
// One wave handles 16 elements (2 nodes x 8 batches).
//   A (16x4): col K=0 = a[m] = agg + self-loop; other cols 0
//   B (4x16): row K=0 = W1[c] (c<8), else 0
//   C (16x16): C[m,c] = b1[c] (c<8)
//   D[m,c] = a[m]*W1[c] + b1[c];  y[m] = sum_c relu(D[m,c])*W2[c]
// Reduction over c = cross-lane xor-shuffle within 16-lane halves.
__global__ void k_transform_wmma(const float* __restrict__ aggt,
                                 const float* __restrict__ xt,
                                 const float* __restrict__ dis,
                                 const float* __restrict__ W1,
                                 const float* __restrict__ b1,
                                 const float* __restrict__ W2,
                                 float* __restrict__ yt, int N) {
  const int elems = N * BATCH;
  const int lane  = threadIdx.x & 31;
  const int wave  = (blockIdx.x * blockDim.x + threadIdx.x) >> 5;
  const int base  = wave * 16;
  if (base >= elems) return;  // wave-uniform exit; EXEC all-1 below

  const int m    = lane & 15;             // row index within tile
  int idx        = base + m;
  int idxc       = idx < elems ? idx : elems - 1;
  int node       = idxc >> 3;
  float dn       = dis[node];
  float aval     = aggt[idxc] + dn * dn * xt[idxc];   // add self-loop term
  float lowMask  = (lane < 16) ? 1.0f : 0.0f;
  float valid    = (idx < elems) ? 1.0f : 0.0f;

  v2f_t A; A.x = aval * lowMask * valid; A.y = 0.0f;          // K=0 col only
  float w1v = (m < 8) ? W1[m] : 0.0f;
  v2f_t Bq; Bq.x = w1v * lowMask; Bq.y = 0.0f;                // K=0 row only
  float b1v = (m < 8) ? b1[m] : 0.0f;
  v8f_t C;
#pragma unroll
  for (int r = 0; r < 8; ++r) C[r] = b1v;                     // C[m,c] = b1[c]

  v8f_t D = __builtin_amdgcn_wmma_f32_16x16x4_f32(
      false, A, false, Bq, (short)0, C, false, false);

  float w2v = (m < 8) ? W2[m] : 0.0f;
  float acc[8];
#pragma unroll
  for (int r = 0; r < 8; ++r) {
    float v = D[r] > 0.0f ? D[r] : 0.0f;   // ReLU
    acc[r] = v * w2v;                      // weight by W2[col]
  }
  // reduce over columns (lanes) inside each 16-lane half
#pragma unroll
  for (int s = 1; s < 16; s <<= 1) {
#pragma unroll
    for (int r = 0; r < 8; ++r) acc[r] += __shfl_xor(acc[r], s, 32);
  }
  // lanes 0-15 now hold y[base+0..7] in acc[0..7]; lanes 16-31 hold y[base+8..15]
  if (lane == 0 || lane == 16) {
    int off = base + ((lane == 16) ? 8 : 0);
#pragma unroll
    for (int r = 0; r < 8; ++r)
      if (off + r < elems) yt[off + r] = acc[r];
  }
}

// --- finalize: out[b*N+n] = sigmoid(acc + dis^2*y + b2) ---------------------
__global__ void k_finalize(const float* __restrict__ out2t,
                           const float* __restrict__ yt,
                           const float* __restrict__ dis,
                           const float* __restrict__ b2,
                           float* __restrict__ out, int N) {
  int t = blockIdx.x * blockDim.x + threadIdx.x;
  if (t >= N * BATCH) return;
  int b = t / N, n = t - b * N;           // contiguous writes to out
  int idx = n * BATCH + b;
  float dn = dis[n];
  float v = out2t[idx] + dn * dn * yt[idx] + b2[0];
  out[t] = 1.0f / (1.0f + expf(-v));
}

extern "C" void kernel_launch(void* const* d_in, const int* in_sizes, int n_in,
                              void* d_out, int out_size, void* d_ws, size_t ws_size,
                              hipStream_t stream) {
  const float*     x   = (const float*)d_in[0];      // [B,N,1]
  const float*     W1  = (const float*)d_in[1];      // [1,8]
  const float*     b1  = (const float*)d_in[2];      // [8]
  const float*     W2  = (const float*)d_in[3];      // [8,1]
  const float*     b2  = (const float*)d_in[4];      // [1]
  const long long* ei  = (const long long*)d_in[5];  // [2,E] int64

  const int N = in_sizes[0] / BATCH;
  const int E = in_sizes[5] / 2;

  // workspace layout (floats)
  float* ws    = (float*)d_ws;
  float* deg   = ws;                       // [N]
  float* dis   = ws + (long long)N;        // [N]
  float* xt    = ws + (long long)2  * N;   // [N][8]
  float* aggt  = ws + (long long)10 * N;   // [N][8]
  float* yt    = ws + (long long)18 * N;   // [N][8]
  float* out2t = ws + (long long)26 * N;   // [N][8]
  const long long totalWs = (long long)34 * N;

  const int T = 256;
  k_zero<<<2048, T, 0, stream>>>(ws, totalWs);
  k_transpose<<<(N * BATCH + T - 1) / T, T, 0, stream>>>(x, xt, N);
  k_deg<<<(E + T - 1) / T, T, 0, stream>>>(ei, deg, E);
  k_rsqrt<<<(N + T - 1) / T, T, 0, stream>>>(deg, dis, N);
  // layer 1 aggregation (scalar per batch, 8 atomics/edge)
  k_edge_scatter<<<(E + T - 1) / T, T, 0, stream>>>(ei, dis, xt, aggt, E);
  // per-node MLP via WMMA: 16 elements per wave
  {
    int tiles = (N * BATCH + 15) / 16;           // waves needed
    int waves_per_block = T / 32;
    int blocks = (tiles + waves_per_block - 1) / waves_per_block;
    k_transform_wmma<<<blocks, T, 0, stream>>>(aggt, xt, dis, W1, b1, W2, yt, N);
  }
  // layer 2 aggregation
  k_edge_scatter<<<(E + T - 1) / T, T, 0, stream>>>(ei, dis, yt, out2t, E);
  k_finalize<<<(N * BATCH + T - 1) / T, T, 0, stream>>>(out2t, yt, dis, b2,
                                                        (float*)d_out, N);
}